// LSTMMultivariateRegressorPerRegion_86071144612203
// MI455X (gfx1250) — compile-verified
//
#include <hip/hip_runtime.h>

typedef __attribute__((ext_vector_type(2))) float v2f;
typedef __attribute__((ext_vector_type(8))) float v8f;

#define T_STEPS  2048
#define BATCH    256
#define IN_DIM   64
#define H0       16
#define H1       32
#define H2       16
#define OUT_DIM  8
#define NTHREADS 128   // 4 waves of 32

__device__ __forceinline__ float sigmoidf_(float x) { return 1.0f / (1.0f + __expf(-x)); }

// A fragment for V_WMMA_F32_16X16X4_F32: A is 16x4 (MxK), row-major source, stride lda.
// Lanes 0-15: M=lane, holds K = kbase+{0,1}; lanes 16-31: M=lane-16, K = kbase+{2,3}.
__device__ __forceinline__ v2f load_afrag(const float* base, int lda, int kbase, int lane) {
    int m = lane & 15;
    int k = kbase + ((lane >> 4) << 1);
    const float* p = base + m * lda + k;
    v2f a;
    a.x = p[0];
    a.y = p[1];
    return a;
}

// B fragment: B is 4x16 (KxN), B[k,n] = W[n,k] with W row-major [N_total, K_total], stride ldw.
// Lanes 0-15: N=lane, K = kbase+{0,1}; lanes 16-31: N=lane-16, K = kbase+{2,3}.
__device__ __forceinline__ v2f load_bfrag(const float* W, int ldw, int nbase, int kbase, int lane) {
    int n = nbase + (lane & 15);
    int k = kbase + ((lane >> 4) << 1);
    const float* p = W + n * ldw + k;
    v2f b;
    b.x = p[0];
    b.y = p[1];
    return b;
}

__device__ __forceinline__ v8f wmma4(v2f a, v2f b, v8f c) {
    // (neg_a, A, neg_b, B, c_mod, C, reuse_a, reuse_b)
    return __builtin_amdgcn_wmma_f32_16x16x4_f32(false, a, false, b, (short)0, c, false, false);
}

// D/C tile 16x16 f32: VGPR j -> row j (lanes 0-15) / row j+8 (lanes 16-31), col = lane&15.
__device__ __forceinline__ void store_dtile(float* g, int ldg, int nbase, int lane, v8f d) {
    int col   = nbase + (lane & 15);
    int rbase = (lane >> 4) << 3;
#pragma unroll
    for (int j = 0; j < 8; ++j)
        g[(rbase + j) * ldg + col] = d[j];
}

template <int H>
__device__ __forceinline__ void cell_update(const float* g, float* c, float* h, int tid) {
    const int fourH = 4 * H;
#pragma unroll
    for (int idx = tid; idx < 16 * H; idx += NTHREADS) {
        int m = idx / H;
        int n = idx % H;
        float vi = g[m * fourH + n];             // gate i
        float vf = g[m * fourH + H + n];         // gate f
        float vg = g[m * fourH + 2 * H + n];     // gate g
        float vo = g[m * fourH + 3 * H + n];     // gate o
        float cc = c[idx];
        float cn = sigmoidf_(vf) * cc + sigmoidf_(vi) * tanhf(vg);
        c[idx] = cn;
        h[idx] = sigmoidf_(vo) * tanhf(cn);
    }
}

__global__ __launch_bounds__(NTHREADS) void lstm3_scan_kernel(
    const float* __restrict__ x,
    const float* __restrict__ Wih0, const float* __restrict__ Whh0,
    const float* __restrict__ bih0, const float* __restrict__ bhh0,
    const float* __restrict__ Wih1, const float* __restrict__ Whh1,
    const float* __restrict__ bih1, const float* __restrict__ bhh1,
    const float* __restrict__ Wih2, const float* __restrict__ Whh2,
    const float* __restrict__ bih2, const float* __restrict__ bhh2,
    const float* __restrict__ Wl,   const float* __restrict__ bl,
    float* __restrict__ out)
{
    __shared__ float sG[16 * 4 * H1];                 // gate scratch, max 16x128
    __shared__ float sH0[16 * H0], sC0[16 * H0];
    __shared__ float sH1[16 * H1], sC1[16 * H1];
    __shared__ float sH2[16 * H2], sC2[16 * H2];
    __shared__ float sWl[OUT_DIM * H2];
    __shared__ float sBl[OUT_DIM];

    const int tid  = threadIdx.x;
    const int lane = tid & 31;
    const int wave = tid >> 5;
    const int ln   = lane & 15;
    const int b0   = blockIdx.x * 16;                 // batch tile base

    // ---- init state / constants in LDS ----
    for (int i = tid; i < 16 * H0; i += NTHREADS) { sH0[i] = 0.f; sC0[i] = 0.f; sH2[i] = 0.f; sC2[i] = 0.f; }
    for (int i = tid; i < 16 * H1; i += NTHREADS) { sH1[i] = 0.f; sC1[i] = 0.f; }
    for (int i = tid; i < OUT_DIM * H2; i += NTHREADS) sWl[i] = Wl[i];
    if (tid < OUT_DIM) sBl[tid] = bl[tid];

    // ---- preload loop-invariant weight B-fragments into VGPRs ----
    // layer 0: this wave owns gate columns nb0..nb0+15 of gates0[16,64]
    const int nb0 = wave * 16;
    v2f wx0[16], wh0[4];
#pragma unroll
    for (int kk = 0; kk < 16; ++kk) wx0[kk] = load_bfrag(Wih0, IN_DIM, nb0, 4 * kk, lane);
#pragma unroll
    for (int kk = 0; kk < 4;  ++kk) wh0[kk] = load_bfrag(Whh0, H0, nb0, 4 * kk, lane);
    const float bias0 = bih0[nb0 + ln] + bhh0[nb0 + ln];

    // layer 1: this wave owns two tiles of gates1[16,128]
    const int nb1a = wave * 32, nb1b = wave * 32 + 16;
    v2f wx1a[4], wx1b[4], wh1a[8], wh1b[8];
#pragma unroll
    for (int kk = 0; kk < 4; ++kk) {
        wx1a[kk] = load_bfrag(Wih1, H0, nb1a, 4 * kk, lane);
        wx1b[kk] = load_bfrag(Wih1, H0, nb1b, 4 * kk, lane);
    }
#pragma unroll
    for (int kk = 0; kk < 8; ++kk) {
        wh1a[kk] = load_bfrag(Whh1, H1, nb1a, 4 * kk, lane);
        wh1b[kk] = load_bfrag(Whh1, H1, nb1b, 4 * kk, lane);
    }
    const float bias1a = bih1[nb1a + ln] + bhh1[nb1a + ln];
    const float bias1b = bih1[nb1b + ln] + bhh1[nb1b + ln];

    // layer 2: one tile of gates2[16,64]
    const int nb2 = wave * 16;
    v2f wx2[8], wh2[4];
#pragma unroll
    for (int kk = 0; kk < 8; ++kk) wx2[kk] = load_bfrag(Wih2, H1, nb2, 4 * kk, lane);
#pragma unroll
    for (int kk = 0; kk < 4; ++kk) wh2[kk] = load_bfrag(Whh2, H2, nb2, 4 * kk, lane);
    const float bias2 = bih2[nb2 + ln] + bhh2[nb2 + ln];

    __syncthreads();

    // ---- sequential scan over T ----
    for (int t = 0; t < T_STEPS; ++t) {
        const float* xt = x + ((size_t)t * BATCH + b0) * IN_DIM;
        if (t + 1 < T_STEPS) {  // pull next step's x tile toward the caches
            const float* xn = xt + (size_t)BATCH * IN_DIM;
            __builtin_prefetch(xn + tid * 8, 0, 0);
        }

        // ===== layer 0: gates0[16,64] = x_t[16,64] @ Wih0^T + h0[16,16] @ Whh0^T + b =====
        v8f acc0;
#pragma unroll
        for (int j = 0; j < 8; ++j) acc0[j] = bias0;
#pragma unroll
        for (int kk = 0; kk < 16; ++kk) {
            v2f a = load_afrag(xt, IN_DIM, 4 * kk, lane);
            acc0 = wmma4(a, wx0[kk], acc0);
        }
#pragma unroll
        for (int kk = 0; kk < 4; ++kk) {
            v2f a = load_afrag(sH0, H0, 4 * kk, lane);
            acc0 = wmma4(a, wh0[kk], acc0);
        }
        store_dtile(sG, 4 * H0, nb0, lane, acc0);
        __syncthreads();
        cell_update<H0>(sG, sC0, sH0, tid);
        __syncthreads();

        // ===== layer 1: gates1[16,128] = h0[16,16] @ Wih1^T + h1[16,32] @ Whh1^T + b =====
        v8f acc1a, acc1b;
#pragma unroll
        for (int j = 0; j < 8; ++j) { acc1a[j] = bias1a; acc1b[j] = bias1b; }
#pragma unroll
        for (int kk = 0; kk < 4; ++kk) {
            v2f a = load_afrag(sH0, H0, 4 * kk, lane);
            acc1a = wmma4(a, wx1a[kk], acc1a);
            acc1b = wmma4(a, wx1b[kk], acc1b);
        }
#pragma unroll
        for (int kk = 0; kk < 8; ++kk) {
            v2f a = load_afrag(sH1, H1, 4 * kk, lane);
            acc1a = wmma4(a, wh1a[kk], acc1a);
            acc1b = wmma4(a, wh1b[kk], acc1b);
        }
        store_dtile(sG, 4 * H1, nb1a, lane, acc1a);
        store_dtile(sG, 4 * H1, nb1b, lane, acc1b);
        __syncthreads();
        cell_update<H1>(sG, sC1, sH1, tid);
        __syncthreads();

        // ===== layer 2: gates2[16,64] = h1[16,32] @ Wih2^T + h2[16,16] @ Whh2^T + b =====
        v8f acc2;
#pragma unroll
        for (int j = 0; j < 8; ++j) acc2[j] = bias2;
#pragma unroll
        for (int kk = 0; kk < 8; ++kk) {
            v2f a = load_afrag(sH1, H1, 4 * kk, lane);
            acc2 = wmma4(a, wx2[kk], acc2);
        }
#pragma unroll
        for (int kk = 0; kk < 4; ++kk) {
            v2f a = load_afrag(sH2, H2, 4 * kk, lane);
            acc2 = wmma4(a, wh2[kk], acc2);
        }
        store_dtile(sG, 4 * H2, nb2, lane, acc2);
        __syncthreads();
        cell_update<H2>(sG, sC2, sH2, tid);
        __syncthreads();

        // ===== output projection: pred[16,8] = h2 @ Wl^T + bl, one element per thread =====
        {
            int m = tid >> 3;        // 0..15
            int o = tid & 7;         // 0..7
            float acc = sBl[o];
#pragma unroll
            for (int k = 0; k < H2; ++k) acc += sH2[m * H2 + k] * sWl[o * H2 + k];
            out[(((size_t)(b0 + m)) * T_STEPS + t) * OUT_DIM + o] = acc;
        }
        // no barrier needed: next write to sH2/sG is separated by barriers above
    }
}

extern "C" void kernel_launch(void* const* d_in, const int* in_sizes, int n_in,
                              void* d_out, int out_size, void* d_ws, size_t ws_size,
                              hipStream_t stream) {
    (void)in_sizes; (void)n_in; (void)out_size; (void)d_ws; (void)ws_size;
    const float* x    = (const float*)d_in[0];
    const float* Wih0 = (const float*)d_in[1];
    const float* Whh0 = (const float*)d_in[2];
    const float* bih0 = (const float*)d_in[3];
    const float* bhh0 = (const float*)d_in[4];
    const float* Wih1 = (const float*)d_in[5];
    const float* Whh1 = (const float*)d_in[6];
    const float* bih1 = (const float*)d_in[7];
    const float* bhh1 = (const float*)d_in[8];
    const float* Wih2 = (const float*)d_in[9];
    const float* Whh2 = (const float*)d_in[10];
    const float* bih2 = (const float*)d_in[11];
    const float* bhh2 = (const float*)d_in[12];
    const float* Wl   = (const float*)d_in[13];
    const float* bl   = (const float*)d_in[14];
    float* out = (float*)d_out;

    lstm3_scan_kernel<<<BATCH / 16, NTHREADS, 0, stream>>>(
        x, Wih0, Whh0, bih0, bhh0,
        Wih1, Whh1, bih1, bhh1,
        Wih2, Whh2, bih2, bhh2,
        Wl, bl, out);
}